// AttentionBlock_58445914964395
// MI455X (gfx1250) — compile-verified
//
#include <hip/hip_runtime.h>
#include <math.h>

// Problem constants (B,C,H,W) = (2,256,64,64), 8 groups, 4 heads, d=64
#define BB   2
#define CC   256
#define LL   4096          // H*W
#define CPG  32            // channels per group (256/8)
#define NHE  4
#define DD   64
#define EPSV 1e-5f

typedef __attribute__((ext_vector_type(2))) float v2f;
typedef __attribute__((ext_vector_type(8))) float v8f;

// fp32 WMMA: D(16x16) = A(16x4) * B(4x16) + C, exact fp32 accumulate
static __device__ __forceinline__ v8f wmma4(v2f a, v2f b, v8f c) {
  return __builtin_amdgcn_wmma_f32_16x16x4_f32(
      /*neg_a=*/false, a, /*neg_b=*/false, b,
      /*c_mod=*/(short)0, c, /*reuse_a=*/false, /*reuse_b=*/false);
}

// ---------------------------------------------------------------------------
// Kernel 1: per-(batch,group) mean / rstd. Group channels are contiguous, so
// each of the 16 blocks reduces one contiguous 512KB slab.
// ---------------------------------------------------------------------------
__global__ void gn_stats(const float* __restrict__ x, float* __restrict__ stats) {
  const int bg = blockIdx.x;                 // 0..15  (b*8 + g)
  const float* p = x + (size_t)bg * CPG * LL;
  float s = 0.f, s2 = 0.f;
  for (int i = threadIdx.x; i < CPG * LL; i += blockDim.x) {
    float v = p[i];
    s  += v;
    s2 += v * v;
  }
  __shared__ float sh[256], sh2[256];
  sh[threadIdx.x] = s; sh2[threadIdx.x] = s2;
  __syncthreads();
  for (int off = 128; off > 0; off >>= 1) {
    if (threadIdx.x < off) {
      sh[threadIdx.x]  += sh[threadIdx.x + off];
      sh2[threadIdx.x] += sh2[threadIdx.x + off];
    }
    __syncthreads();
  }
  if (threadIdx.x == 0) {
    const float invn = 1.0f / (float)(CPG * LL);
    float mean = sh[0] * invn;
    float var  = sh2[0] * invn - mean * mean;
    stats[2 * bg]     = mean;
    stats[2 * bg + 1] = rsqrtf(var + EPSV);
  }
}

// ---------------------------------------------------------------------------
// Kernel 2: apply GroupNorm -> xn (kept for residual + as GEMM operand)
// ---------------------------------------------------------------------------
__global__ void gn_apply(const float* __restrict__ x,
                         const float* __restrict__ gamma,
                         const float* __restrict__ beta,
                         const float* __restrict__ stats,
                         float* __restrict__ xn) {
  size_t i = (size_t)blockIdx.x * blockDim.x + threadIdx.x;  // < 2^21
  int c  = (int)((i / LL) % CC);
  int bg = (int)(i / ((size_t)CPG * LL));
  float mean = stats[2 * bg];
  float rstd = stats[2 * bg + 1];
  xn[i] = (x[i] - mean) * rstd * gamma[c] + beta[c];
}

// ---------------------------------------------------------------------------
// Kernel 3/5: D[b][o][p] = sum_c W[o][c] * X[b][c][p]  (+ bias[o] + res[b][o][p])
// One wave computes one 16x16 output tile via 64 x V_WMMA_F32_16X16X4_F32.
// Block = 8 waves covering 128 columns. grid = (LL/128, M/16, B)
// ---------------------------------------------------------------------------
__global__ void gemm_wc(const float* __restrict__ W, const float* __restrict__ X,
                        float* __restrict__ Dst, const float* __restrict__ bias,
                        const float* __restrict__ res, int M, int K) {
  const int b    = blockIdx.z;
  const int wave = threadIdx.x >> 5;
  const int lane = threadIdx.x & 31;
  const int hi   = lane >> 4;          // half-wave select (K split for f32 frags)
  const int ln   = lane & 15;
  const int M0   = blockIdx.y * 16;
  const int N0   = blockIdx.x * 128 + wave * 16;
  const float* Xb = X + (size_t)b * K * LL;

  v8f acc = {};
  for (int k = 0; k < K; k += 4) {
    const int kk = k + 2 * hi;
    // A frag: lane ln holds row M0+ln, K = kk,kk+1 (contiguous -> b64 load)
    v2f a = *(const v2f*)(W + (size_t)(M0 + ln) * K + kk);
    // B frag: lane ln holds col N0+ln, rows kk,kk+1
    v2f bf;
    bf.x = Xb[(size_t)kk * LL + N0 + ln];
    bf.y = Xb[(size_t)(kk + 1) * LL + N0 + ln];
    acc = wmma4(a, bf, acc);
  }

  float* Db = Dst + (size_t)b * M * LL;
#pragma unroll
  for (int e = 0; e < 8; ++e) {
    const int row = M0 + e + 8 * hi;   // C/D layout: row = elem + 8*(lane>=16)
    const int col = N0 + ln;
    float v = acc[e];
    if (bias) v += bias[row];
    if (res)  v += res[(size_t)b * M * LL + (size_t)row * LL + col];
    Db[(size_t)row * LL + col] = v;
  }
}

// ---------------------------------------------------------------------------
// Kernel 4: flash attention per (b,h). One wave per 16-query tile.
// S = (Q^T K) * d^-0.5 via WMMA (K-dim = 64 channels, 16 steps of 4)
// online softmax in C/D-register layout, P relayout via 1KB LDS,
// O(16x64) += P @ V^T via WMMA (4 c-tiles x 4 K-steps).
// ---------------------------------------------------------------------------
__global__ void attn_kernel(const float* __restrict__ qkv, float* __restrict__ out) {
  __shared__ float pbuf[16 * 16];
  const int bh = blockIdx.y;                // 0..7
  const int b  = bh >> 2, h = bh & 3;
  const int I0 = blockIdx.x * 16;
  const int lane = threadIdx.x & 31;
  const int hi = lane >> 4, ln = lane & 15;

  const size_t bstr = (size_t)3 * CC * LL;
  const float* Qm = qkv + (size_t)b * bstr + (size_t)(h * DD) * LL;
  const float* Km = Qm + (size_t)CC * LL;   // K rows start at o=256
  const float* Vm = Km + (size_t)CC * LL;   // V rows start at o=512

  // Preload all 16 Q A-fragments (A[m][kc] = Q[kc][I0+m]) into registers.
  v2f Aq[16];
#pragma unroll
  for (int t = 0; t < 16; ++t) {
    const int kc = 4 * t + 2 * hi;
    v2f a;
    a.x = Qm[(size_t)kc * LL + I0 + ln];
    a.y = Qm[(size_t)(kc + 1) * LL + I0 + ln];
    Aq[t] = a;
  }

  v8f O0 = {}, O1 = {}, O2 = {}, O3 = {};
  float m8[8], l8[8];
#pragma unroll
  for (int e = 0; e < 8; ++e) { m8[e] = -3.0e38f; l8[e] = 0.f; }
  const float scale = 0.125f;               // d^-0.5, d=64

  for (int jt = 0; jt < LL / 16; ++jt) {
    const int J0 = jt * 16;

    // ---- S tile = Q^T K  (16 WMMA, K=64) ----
    v8f s = {};
#pragma unroll
    for (int t = 0; t < 16; ++t) {
      const int kc = 4 * t + 2 * hi;
      v2f bk;
      bk.x = Km[(size_t)kc * LL + J0 + ln];
      bk.y = Km[(size_t)(kc + 1) * LL + J0 + ln];
      s = wmma4(Aq[t], bk, s);
    }

    // ---- online softmax (rows live across 16-lane halves) ----
    float fac[8];
#pragma unroll
    for (int e = 0; e < 8; ++e) {
      float v = s[e] * scale;
      float r = v;
      r = fmaxf(r, __shfl_xor(r, 1, 16));
      r = fmaxf(r, __shfl_xor(r, 2, 16));
      r = fmaxf(r, __shfl_xor(r, 4, 16));
      r = fmaxf(r, __shfl_xor(r, 8, 16));
      const float nm = fmaxf(m8[e], r);
      const float p  = __expf(v - nm);
      float rs = p;
      rs += __shfl_xor(rs, 1, 16);
      rs += __shfl_xor(rs, 2, 16);
      rs += __shfl_xor(rs, 4, 16);
      rs += __shfl_xor(rs, 8, 16);
      const float f = __expf(m8[e] - nm);
      l8[e] = l8[e] * f + rs;
      m8[e] = nm;
      fac[e] = f;
      pbuf[(e + 8 * hi) * 16 + ln] = p;     // D-layout -> row-major in LDS
    }
#pragma unroll
    for (int e = 0; e < 8; ++e) {
      O0[e] *= fac[e]; O1[e] *= fac[e]; O2[e] *= fac[e]; O3[e] *= fac[e];
    }
    __syncthreads();   // single-wave workgroup: S_NOP, but orders LDS use

    // ---- O += P @ V^T  (16 WMMA: 4 K-steps x 4 c-tiles) ----
#pragma unroll
    for (int t2 = 0; t2 < 4; ++t2) {
      const int kk = 4 * t2 + 2 * hi;
      v2f ap;
      ap.x = pbuf[ln * 16 + kk];
      ap.y = pbuf[ln * 16 + kk + 1];
      v2f bv;
      bv = *(const v2f*)(Vm + (size_t)(0  + ln) * LL + J0 + kk);
      O0 = wmma4(ap, bv, O0);
      bv = *(const v2f*)(Vm + (size_t)(16 + ln) * LL + J0 + kk);
      O1 = wmma4(ap, bv, O1);
      bv = *(const v2f*)(Vm + (size_t)(32 + ln) * LL + J0 + kk);
      O2 = wmma4(ap, bv, O2);
      bv = *(const v2f*)(Vm + (size_t)(48 + ln) * LL + J0 + kk);
      O3 = wmma4(ap, bv, O3);
    }
    __syncthreads();
  }

  // ---- finalize: out[b][h*64+c][i] = O[i][c] / l[i] ----
  float* Ob = out + ((size_t)b * CC + h * DD) * LL;
#pragma unroll
  for (int e = 0; e < 8; ++e) {
    const float inv = 1.0f / l8[e];
    const int i = I0 + e + 8 * hi;
    Ob[(size_t)(0  + ln) * LL + i] = O0[e] * inv;
    Ob[(size_t)(16 + ln) * LL + i] = O1[e] * inv;
    Ob[(size_t)(32 + ln) * LL + i] = O2[e] * inv;
    Ob[(size_t)(48 + ln) * LL + i] = O3[e] * inv;
  }
}

// ---------------------------------------------------------------------------
extern "C" void kernel_launch(void* const* d_in, const int* in_sizes, int n_in,
                              void* d_out, int out_size, void* d_ws, size_t ws_size,
                              hipStream_t stream) {
  (void)in_sizes; (void)n_in; (void)out_size; (void)ws_size;
  const float* x      = (const float*)d_in[0];
  const float* gamma  = (const float*)d_in[1];
  const float* beta   = (const float*)d_in[2];
  const float* w_qkv  = (const float*)d_in[3];
  const float* w_proj = (const float*)d_in[4];
  const float* b_proj = (const float*)d_in[5];
  float* out = (float*)d_out;

  // workspace layout (floats): stats[32] | xn[2M] | qkv[6M] | attnout[2M] ~ 40MB
  float* wsf     = (float*)d_ws;
  float* stats   = wsf;
  float* xn      = wsf + 64;
  float* qkv     = xn  + (size_t)BB * CC * LL;
  float* attnout = qkv + (size_t)BB * 3 * CC * LL;

  gn_stats<<<BB * 8, 256, 0, stream>>>(x, stats);
  gn_apply<<<(BB * CC * LL) / 256, 256, 0, stream>>>(x, gamma, beta, stats, xn);
  // QKV: M=768, K=256
  gemm_wc<<<dim3(LL / 128, 768 / 16, BB), 256, 0, stream>>>(
      w_qkv, xn, qkv, nullptr, nullptr, 768, CC);
  // attention: one 16-query tile per wave, grid (256 tiles, B*NH)
  attn_kernel<<<dim3(LL / 16, BB * NHE), 32, 0, stream>>>(qkv, attnout);
  // proj + bias + residual: M=256, K=256
  gemm_wc<<<dim3(LL / 128, CC / 16, BB), 256, 0, stream>>>(
      w_proj, attnout, out, b_proj, xn, CC, CC);
}